// ArgmaxCodeDecoder_66709432041859
// MI455X (gfx1250) — compile-verified
//
#include <hip/hip_runtime.h>

#define NL 28
#define HDIM 1024
#define NHEAD 16
#define NKVH 8
#define HD 128
#define KVD 1024   // NKVH*HD
#define DFF 2048
#define VOC 3072
#define SEQ 256
#define RMS_EPS 1e-6f

#define NBLK 256
#define NTHR 256
#define NWAVE 8

typedef __attribute__((ext_vector_type(2))) float v2f;
typedef __attribute__((ext_vector_type(8))) float v8f;

// workspace float layout (after 16-float barrier header)
#define WS_H    0
#define WS_QKV  1024
#define WS_ATTN 5120
#define WS_G    7168
#define WS_U    9216

// ---------------- grid-wide split barrier (persistent kernel) ----------------
__device__ __forceinline__ void gbar(unsigned* cnt, unsigned* gen) {
    __syncthreads();
    __threadfence();
    if (threadIdx.x == 0) {
        unsigned g = __hip_atomic_load(gen, __ATOMIC_RELAXED, __HIP_MEMORY_SCOPE_AGENT);
        unsigned a = __hip_atomic_fetch_add(cnt, 1u, __ATOMIC_ACQ_REL, __HIP_MEMORY_SCOPE_AGENT);
        if (a == (unsigned)(NBLK - 1)) {
            __hip_atomic_store(cnt, 0u, __ATOMIC_RELAXED, __HIP_MEMORY_SCOPE_AGENT);
            __threadfence();
            __hip_atomic_fetch_add(gen, 1u, __ATOMIC_ACQ_REL, __HIP_MEMORY_SCOPE_AGENT);
        } else {
            while (__hip_atomic_load(gen, __ATOMIC_ACQUIRE, __HIP_MEMORY_SCOPE_AGENT) == g) {
                __builtin_amdgcn_s_sleep(1);
            }
        }
    }
    __threadfence();
    __syncthreads();
}

// ---------------- block reductions (wave32) ----------------
__device__ __forceinline__ float bsum(float v, float* red) {
    const int tid = threadIdx.x, lane = tid & 31, w = tid >> 5;
#pragma unroll
    for (int o = 16; o > 0; o >>= 1) v += __shfl_down(v, o, 32);
    __syncthreads();
    if (lane == 0) red[w] = v;
    __syncthreads();
    if (tid == 0) {
        float s = 0.f;
#pragma unroll
        for (int i = 0; i < NWAVE; ++i) s += red[i];
        red[NWAVE] = s;
    }
    __syncthreads();
    return red[NWAVE];
}

__device__ __forceinline__ float bmax(float v, float* red) {
    const int tid = threadIdx.x, lane = tid & 31, w = tid >> 5;
#pragma unroll
    for (int o = 16; o > 0; o >>= 1) v = fmaxf(v, __shfl_down(v, o, 32));
    __syncthreads();
    if (lane == 0) red[w] = v;
    __syncthreads();
    if (tid == 0) {
        float s = red[0];
#pragma unroll
        for (int i = 1; i < NWAVE; ++i) s = fmaxf(s, red[i]);
        red[NWAVE] = s;
    }
    __syncthreads();
    return red[NWAVE];
}

// RMS-norm a 1024-vector from global scratch into LDS (redundant per block)
__device__ __forceinline__ void prep_rms1024(const float* __restrict__ hsrc,
                                             const float* __restrict__ w,
                                             float* xs, float* red) {
    const int tid = threadIdx.x;
    float4 h4 = ((const float4*)hsrc)[tid];
    float ss = h4.x * h4.x + h4.y * h4.y + h4.z * h4.z + h4.w * h4.w;
    ss = bsum(ss, red);
    float sc = rsqrtf(ss * (1.0f / (float)HDIM) + RMS_EPS);
    float4 w4 = ((const float4*)w)[tid];
    float4 o;
    o.x = h4.x * sc * w4.x; o.y = h4.y * sc * w4.y;
    o.z = h4.z * sc * w4.z; o.w = h4.w * sc * w4.w;
    ((float4*)xs)[tid] = o;
    __syncthreads();
}

// ---------------- WMMA fp32 matvec: 16 rows/block-group, 8-way K split ----------
// A (16x4 f32): lanes 0-15 hold {K0,K1} of row M=lane; lanes 16-31 hold {K2,K3}.
// B (4x16 f32): replicate x[k] across the 16 lanes (columns) of each half-wave,
//   VGPR0/1 hold {K0,K1} (lanes<16) and {K2,K3} (lanes>=16) -> all C columns equal.
// C/D: VGPR v = row v (lanes 0-15, N=lane) / row v+8 (lanes 16-31) -> read N=0
//   from lane 0 and lane 16.
// Weights are streamed once -> non-temporal loads keep the 192MB L2 for
// KV cache + activations.
template <class RP, class EP>
__device__ __forceinline__ void matvec16(int G, int K, const float* xs,
                                         RP rowp, EP epi, float (*slots)[16]) {
    const int tid  = threadIdx.x;
    const int lane = tid & 31;
    const int wave = tid >> 5;
    const int Kc = K >> 3;            // 8-way K split across the block's waves
    const int kbase = wave * Kc;
    const int o = (lane >> 4) << 1;   // 0 for lanes<16, 2 for lanes>=16
    for (int g = blockIdx.x; g < G; g += NBLK) {
        const float* wrow = rowp(g * 16 + (lane & 15));
        const float* wp = wrow + kbase + o;
        v8f c = {};
#pragma unroll 16
        for (int kk = 0; kk < Kc; kk += 4) {
            v2f b;
            b.x = xs[kbase + kk + o];
            b.y = xs[kbase + kk + o + 1];
            v2f a = __builtin_nontemporal_load((const v2f*)(wp + kk));
            c = __builtin_amdgcn_wmma_f32_16x16x4_f32(
                    false, a, false, b, (short)0, c, false, false);
        }
        __syncthreads();
        if (lane == 0 || lane == 16) {
            float* sl = &slots[wave][(lane >> 4) << 3];
#pragma unroll
            for (int v = 0; v < 8; ++v) sl[v] = c[v];
        }
        __syncthreads();
        if (wave == 0 && lane < 16) {
            float s = 0.f;
#pragma unroll
            for (int w = 0; w < NWAVE; ++w) s += slots[w][lane];  // fixed order: deterministic
            epi(g * 16 + lane, s);
        }
    }
    __syncthreads();
}

// ---------------- fused persistent decode kernel ----------------
__global__ __launch_bounds__(NTHR) void decode_fused(
    const float* __restrict__ embeds, const int* __restrict__ cachelen,
    const float* __restrict__ padmask, const float* __restrict__ updmask,
    const float* __restrict__ keycache, const float* __restrict__ valcache,
    const float* __restrict__ invfreq, const float* __restrict__ ln1w,
    const float* __restrict__ ln2w, const float* __restrict__ qnw,
    const float* __restrict__ knw, const float* __restrict__ Wq,
    const float* __restrict__ Wk, const float* __restrict__ Wv,
    const float* __restrict__ Wo, const float* __restrict__ Wg,
    const float* __restrict__ Wu, const float* __restrict__ Wd,
    const float* __restrict__ normw, const float* __restrict__ Wh,
    float* __restrict__ out, float* __restrict__ wsraw)
{
    __shared__ float xs[DFF];
    __shared__ float slots[NWAVE][16];
    __shared__ float red[NWAVE + 1];
    __shared__ float aq[HD], ak[HD], av[HD], ap[SEQ];

    unsigned* cnt = (unsigned*)wsraw;
    unsigned* gen = cnt + 1;
    float* ws    = wsraw + 16;
    float* hbuf  = ws + WS_H;
    float* qkv   = ws + WS_QKV;
    float* attnb = ws + WS_ATTN;
    float* gbuf  = ws + WS_G;
    float* ubuf  = ws + WS_U;
    float* outnk = out + 4096;
    float* outnv = out + 4096 + (size_t)NL * KVD * SEQ;

    const int tid = threadIdx.x;
    const int pos = cachelen[0];
    const float posf = (float)pos;

    if (blockIdx.x == 0) ((float4*)hbuf)[tid] = ((const float4*)embeds)[tid];
    gbar(cnt, gen);

    for (int l = 0; l < NL; ++l) {
        // ---- stage 1: rms(ln1) + fused QKV matvec (4096 rows x 1024) ----
        prep_rms1024(hbuf, ln1w + (size_t)l * HDIM, xs, red);
        {
            const float* wq = Wq + (size_t)l * 2048 * HDIM;
            const float* wk = Wk + (size_t)l * 1024 * HDIM;
            const float* wv = Wv + (size_t)l * 1024 * HDIM;
            auto rowp = [&](int r) -> const float* {
                return (r < 2048) ? wq + (size_t)r * HDIM
                     : (r < 3072) ? wk + (size_t)(r - 2048) * HDIM
                                  : wv + (size_t)(r - 3072) * HDIM;
            };
            auto epi = [&](int r, float v) { qkv[r] = v; };
            matvec16(256, HDIM, xs, rowp, epi, slots);
        }
        gbar(cnt, gen);

        // ---- stage 2: qk-norm + RoPE + attention (16 head blocks) + KV copy ----
        if (blockIdx.x < NHEAD) {
            const int hh = blockIdx.x;
            const int kvh = hh >> 1;                  // n_rep = 2
            const float* qhp = qkv + hh * HD;
            const float* khp = qkv + 2048 + kvh * HD;
            const float* vhp = qkv + 3072 + kvh * HD;
            float qv = (tid < HD) ? qhp[tid] : 0.f;
            float kv = (tid < HD) ? khp[tid] : 0.f;
            float sq = bsum(qv * qv, red);
            float sk = bsum(kv * kv, red);
            float qsc = rsqrtf(sq * (1.0f / (float)HD) + RMS_EPS);
            float ksc = rsqrtf(sk * (1.0f / (float)HD) + RMS_EPS);
            if (tid < HD) {
                aq[tid] = qv * qsc * qnw[(size_t)l * HD + tid];
                ak[tid] = kv * ksc * knw[(size_t)l * HD + tid];
                av[tid] = vhp[tid];
            }
            __syncthreads();
            float qr = 0.f, kr = 0.f;
            if (tid < HD) {
                float fr = invfreq[tid & 63] * posf;
                float cv = cosf(fr), sv = sinf(fr);
                float sgn = (tid < 64) ? -1.f : 1.f;
                qr = aq[tid] * cv + sgn * aq[tid ^ 64] * sv;
                kr = ak[tid] * cv + sgn * ak[tid ^ 64] * sv;
            }
            __syncthreads();
            if (tid < HD) { aq[tid] = qr; ak[tid] = kr; }
            __syncthreads();
            // write the new cache column (one writer per kv head)
            if ((hh & 1) == 0 && tid < HD) {
                size_t cbase = ((size_t)l * KVD + (size_t)kvh * HD + tid) * SEQ + pos;
                outnk[cbase] = ak[tid];
                outnv[cbase] = av[tid];
            }
            // scores for all 256 positions (thread = position)
            const float* kcb = keycache + ((size_t)l * KVD + (size_t)kvh * HD) * SEQ;
            float sc = 0.f;
            if (tid == pos) {
#pragma unroll 4
                for (int d = 0; d < HD; ++d) sc += aq[d] * ak[d];
            } else {
#pragma unroll 4
                for (int d = 0; d < HD; ++d) sc += aq[d] * kcb[d * SEQ + tid];
            }
            sc = sc * 0.08838834764831845f + padmask[tid];  // 1/sqrt(128)
            float mx = bmax(sc, red);
            float e = expf(sc - mx);
            float se = bsum(e, red);
            ap[tid] = e / se;
            __syncthreads();
            // probs . V (thread = head dim)
            const float* vcb = valcache + ((size_t)l * KVD + (size_t)kvh * HD) * SEQ;
            if (tid < HD) {
                const float* vrow = vcb + tid * SEQ;
                float a = 0.f;
#pragma unroll 4
                for (int s2 = 0; s2 < SEQ; ++s2) a += ap[s2] * vrow[s2];
                a += ap[pos] * (av[tid] - vrow[pos]);  // substitute new column
                attnb[hh * HD + tid] = a;
            }
        } else {
            // stream this layer's KV cache (all s != pos) into outputs;
            // read-once/write-once -> non-temporal both ways
            const int cb = blockIdx.x - NHEAD;
            const int NCB = NBLK - NHEAD;
            const size_t lb = (size_t)l * (KVD * SEQ);
            for (int idx = cb * NTHR + tid; idx < KVD * SEQ; idx += NCB * NTHR) {
                if ((idx & (SEQ - 1)) != pos) {
                    __builtin_nontemporal_store(
                        __builtin_nontemporal_load(keycache + lb + idx), outnk + lb + idx);
                    __builtin_nontemporal_store(
                        __builtin_nontemporal_load(valcache + lb + idx), outnv + lb + idx);
                }
            }
        }
        gbar(cnt, gen);

        // ---- stage 3: h += Wo @ attn (1024 rows x 2048) ----
        {
            ((float4*)xs)[tid]       = ((const float4*)attnb)[tid];
            ((float4*)xs)[tid + 256] = ((const float4*)attnb)[tid + 256];
            __syncthreads();
            const float* wo = Wo + (size_t)l * HDIM * DFF;
            auto rowp = [&](int r) { return wo + (size_t)r * DFF; };
            auto epi = [&](int r, float v) { hbuf[r] += v; };
            matvec16(64, DFF, xs, rowp, epi, slots);
        }
        gbar(cnt, gen);

        // ---- stage 4: rms(ln2) + gate/up matvecs (4096 rows x 1024) ----
        prep_rms1024(hbuf, ln2w + (size_t)l * HDIM, xs, red);
        {
            const float* wg = Wg + (size_t)l * DFF * HDIM;
            const float* wu = Wu + (size_t)l * DFF * HDIM;
            auto rowp = [&](int r) {
                return (r < DFF) ? wg + (size_t)r * HDIM : wu + (size_t)(r - DFF) * HDIM;
            };
            auto epi = [&](int r, float v) {
                if (r < DFF) gbuf[r] = v; else ubuf[r - DFF] = v;
            };
            matvec16(256, HDIM, xs, rowp, epi, slots);
        }
        gbar(cnt, gen);

        // ---- stage 5: ff = silu(g)*u ; h += Wd @ ff (1024 rows x 2048) ----
        {
#pragma unroll
            for (int rep = 0; rep < 2; ++rep) {
                int i = tid + rep * 256;
                float4 g4 = ((const float4*)gbuf)[i];
                float4 u4 = ((const float4*)ubuf)[i];
                float4 f4;
                f4.x = (g4.x / (1.0f + expf(-g4.x))) * u4.x;
                f4.y = (g4.y / (1.0f + expf(-g4.y))) * u4.y;
                f4.z = (g4.z / (1.0f + expf(-g4.z))) * u4.z;
                f4.w = (g4.w / (1.0f + expf(-g4.w))) * u4.w;
                ((float4*)xs)[i] = f4;
            }
            __syncthreads();
            const float* wd = Wd + (size_t)l * HDIM * DFF;
            auto rowp = [&](int r) { return wd + (size_t)r * DFF; };
            auto epi = [&](int r, float v) { hbuf[r] += v; };
            matvec16(64, DFF, xs, rowp, epi, slots);
        }
        gbar(cnt, gen);
    }

    // ---- final: rms(norm_w) -> hidden_out + logits = Wh @ normed ----
    prep_rms1024(hbuf, normw, xs, red);
    if (blockIdx.x == 0) ((float4*)(out + 3072))[tid] = ((const float4*)xs)[tid];
    {
        auto rowp = [&](int r) { return Wh + (size_t)r * HDIM; };
        auto epi = [&](int r, float v) { out[r] = v; };
        matvec16(192, HDIM, xs, rowp, epi, slots);
    }
}

extern "C" void kernel_launch(void* const* d_in, const int* in_sizes, int n_in,
                              void* d_out, int out_size, void* d_ws, size_t ws_size,
                              hipStream_t stream) {
    (void)in_sizes; (void)n_in; (void)out_size; (void)ws_size;
    // zero the barrier header (counter + generation)
    hipMemsetAsync(d_ws, 0, 256, stream);
    decode_fused<<<NBLK, NTHR, 0, stream>>>(
        (const float*)d_in[0],  (const int*)d_in[1],   (const float*)d_in[2],
        (const float*)d_in[3],  (const float*)d_in[4], (const float*)d_in[5],
        (const float*)d_in[6],  (const float*)d_in[7], (const float*)d_in[8],
        (const float*)d_in[9],  (const float*)d_in[10], (const float*)d_in[11],
        (const float*)d_in[12], (const float*)d_in[13], (const float*)d_in[14],
        (const float*)d_in[15], (const float*)d_in[16], (const float*)d_in[17],
        (const float*)d_in[18], (const float*)d_in[19],
        (float*)d_out, (float*)d_ws);
}